// MambaVisionMixer_71631464562944
// MI455X (gfx1250) — compile-verified
//
#include <hip/hip_runtime.h>
#include <math.h>

typedef __attribute__((ext_vector_type(2))) float v2f;
typedef __attribute__((ext_vector_type(8))) float v8f;
typedef __attribute__((ext_vector_type(4))) int   i4;
typedef __attribute__((address_space(1))) i4* gi4p;   // global int4*
typedef __attribute__((address_space(3))) i4* li4p;   // LDS int4*

#define D_MODEL 512
#define D_HALF  256
#define D_STATE 8
#define DT_RANK 32
#define B_SZ    4
#define L_SEQ   4096
#define M_ROWS  (B_SZ * L_SEQ)          // 16384
#define XDBL_W  (DT_RANK + 2 * D_STATE) // 48
#define CHUNK   64
#define NCHUNK  (L_SEQ / CHUNK)         // 64

// GEMM tiling
#define BM 128            // rows per block (8 waves x 16)
#define BN 64             // cols per block (4 n-subtiles x 16)
#define KC 32             // K chunk staged in LDS
#define BPITCH 36         // LDS row pitch (mult of 4 -> 16B aligned; conflict-free)

#if __has_builtin(__builtin_amdgcn_global_load_async_to_lds_b128)
#define HAVE_ASYNC_LDS 1
#else
#define HAVE_ASYNC_LDS 0
#endif

__device__ __forceinline__ void async_wait_2() {
#if HAVE_ASYNC_LDS
#if __has_builtin(__builtin_amdgcn_s_wait_asynccnt)
    __builtin_amdgcn_s_wait_asynccnt(2);
#else
    asm volatile("s_wait_asynccnt 0x2" ::: "memory");
#endif
#endif
}
__device__ __forceinline__ void async_wait_0() {
#if HAVE_ASYNC_LDS
#if __has_builtin(__builtin_amdgcn_s_wait_asynccnt)
    __builtin_amdgcn_s_wait_asynccnt(0);
#else
    asm volatile("s_wait_asynccnt 0x0" ::: "memory");
#endif
#endif
}

// ---------------------------------------------------------------------------
// fp32 WMMA GEMM:  C[M,N] = A[M,K] * W[N,K]^T   (row-major)
// Block = 256 threads (8 waves) -> 128x64 output tile; wave -> 16x64 (4 accs).
// Weight tile (64 x KC) staged in LDS via async-to-LDS, double buffered.
// epi==1: C = softplus(acc + 2*bias[col])
// ---------------------------------------------------------------------------
__global__ void __launch_bounds__(256)
gemm_f32_wmma(const float* __restrict__ A, int lda,
              const float* __restrict__ W, int ldw,
              float* __restrict__ C, int ldc,
              int M, int N, int K,
              const float* __restrict__ bias, int epi)
{
    __shared__ float Bs[2][BN][BPITCH];

    const int wv   = threadIdx.x >> 5;     // 0..7  : m-subtile
    const int lane = threadIdx.x & 31;
    const int hi   = lane >> 4;            // 0: K0/K1, 1: K2/K3
    const int lr   = lane & 15;
    const int nb64 = blockIdx.x * BN;      // first output col of this block
    const int mrow = blockIdx.y * BM + wv * 16 + lr;

    // A fragment source: lane row = mrow, regs hold K = 2*hi + {0,1}
    const float* Arow = A + (size_t)mrow * lda + (hi << 1);

    // Stage one 64 x KC weight chunk into Bs[buf] (zero-fill rows >= N).
    auto stage = [&](int buf, int kc) {
        #pragma unroll
        for (int i = 0; i < 2; ++i) {                 // 512 float4 / 256 thr
            int idx = threadIdx.x + i * 256;
            int r   = idx >> 3;                       // 8 float4 per row
            int c4  = idx & 7;
            int nrow = nb64 + r;
            float* lp = &Bs[buf][r][c4 << 2];
            if (nrow < N) {
                const float* gp = W + (size_t)nrow * ldw + kc + (c4 << 2);
#if HAVE_ASYNC_LDS
                __builtin_amdgcn_global_load_async_to_lds_b128(
                    (gi4p)gp, (li4p)lp, 0, 0);
#else
                *(float4*)lp = *(const float4*)gp;
#endif
            } else {
                float4 z = make_float4(0.f, 0.f, 0.f, 0.f);
                *(float4*)lp = z;
            }
        }
    };

    v8f acc[4] = {};
    const int nchunks = K / KC;

    stage(0, 0);
    for (int c = 0; c < nchunks; ++c) {
        const int buf = c & 1;
        if (c + 1 < nchunks) { stage(buf ^ 1, (c + 1) * KC); async_wait_2(); }
        else                 { async_wait_0(); }
        __syncthreads();

        const int kc = c * KC;
        #pragma unroll
        for (int kk = 0; kk < KC; kk += 4) {
            v2f a = *(const v2f*)(Arow + kc + kk);
            #pragma unroll
            for (int j = 0; j < 4; ++j) {
                // B frag: regs hold B[k+2hi+{0,1}][n], n = j*16+lr
                v2f b = *(const v2f*)(&Bs[buf][j * 16 + lr][kk + (hi << 1)]);
                acc[j] = __builtin_amdgcn_wmma_f32_16x16x4_f32(
                    false, a, false, b, (short)0, acc[j], false, false);
            }
        }
        __syncthreads();
    }

    // Epilogue: C/D layout -> reg r = row (r + 8*hi), col = lane&15
    #pragma unroll
    for (int j = 0; j < 4; ++j) {
        const int col = nb64 + j * 16 + lr;
        if (col < N) {
            float badd = (epi == 1) ? 2.0f * bias[col] : 0.f;
            #pragma unroll
            for (int r = 0; r < 8; ++r) {
                float v = acc[j][r];
                if (epi == 1) {
                    float x = v + badd;
                    v = (x > 20.f) ? x : log1pf(expf(x));   // softplus
                }
                const int row = blockIdx.y * BM + wv * 16 + hi * 8 + r;
                C[(size_t)row * ldc + col] = v;
            }
        }
    }
}

// ---------------------------------------------------------------------------
// Depthwise conv1d (k=3, SAME) + SiLU for both halves of in_proj output.
// ---------------------------------------------------------------------------
__global__ void conv_silu(const float* __restrict__ P,
                          const float* __restrict__ wv,
                          const float* __restrict__ wh,
                          float* __restrict__ Z,
                          float* __restrict__ Hs)
{
    int idx = blockIdx.x * blockDim.x + threadIdx.x;
    if (idx >= B_SZ * L_SEQ * D_HALF) return;
    int d = idx & (D_HALF - 1);
    int l = (idx >> 8) & (L_SEQ - 1);
    int b = idx >> 20;

    size_t rowm  = (size_t)(b * L_SEQ + l) * D_MODEL;
    size_t rprev = rowm - D_MODEL, rnext = rowm + D_MODEL;
    float vm = (l > 0)         ? P[rprev + d] : 0.f;
    float v0 =                   P[rowm  + d];
    float vp = (l < L_SEQ - 1) ? P[rnext + d] : 0.f;
    float hm = (l > 0)         ? P[rprev + D_HALF + d] : 0.f;
    float h0 =                   P[rowm  + D_HALF + d];
    float hp = (l < L_SEQ - 1) ? P[rnext + D_HALF + d] : 0.f;

    float cv = vm * wv[d * 3] + v0 * wv[d * 3 + 1] + vp * wv[d * 3 + 2];
    float ch = hm * wh[d * 3] + h0 * wh[d * 3 + 1] + hp * wh[d * 3 + 2];
    cv = cv / (1.f + expf(-cv));   // SiLU
    ch = ch / (1.f + expf(-ch));
    Z[rowm + d] = cv;
    Hs[(size_t)(b * L_SEQ + l) * D_HALF + d] = ch;
}

// ---------------------------------------------------------------------------
// Chunked selective scan (linear recurrence x_t = dA_t*x + dBu_t).
// ---------------------------------------------------------------------------
__global__ void scan_pass1(const float* __restrict__ Delta,
                           const float* __restrict__ Hs,
                           const float* __restrict__ Xdbl,
                           const float* __restrict__ A_log,
                           float* __restrict__ Pc, float* __restrict__ Xc)
{
    int idx = blockIdx.x * blockDim.x + threadIdx.x;
    if (idx >= B_SZ * NCHUNK * D_HALF) return;
    int d = idx & (D_HALF - 1);
    int c = (idx >> 8) & (NCHUNK - 1);
    int b = idx >> 14;

    float a[D_STATE], p[D_STATE], x[D_STATE];
    #pragma unroll
    for (int n = 0; n < D_STATE; ++n) {
        a[n] = -expf(A_log[d * D_STATE + n]);
        p[n] = 1.f; x[n] = 0.f;
    }
    int l0 = c * CHUNK;
    for (int t = 0; t < CHUNK; ++t) {
        size_t row = (size_t)b * L_SEQ + l0 + t;
        float dl = Delta[row * D_HALF + d];
        float u  = Hs[row * D_HALF + d];
        const float* bv = Xdbl + row * XDBL_W + DT_RANK;
        float du = dl * u;
        #pragma unroll
        for (int n = 0; n < D_STATE; ++n) {
            float dA = expf(dl * a[n]);
            x[n] = dA * x[n] + du * bv[n];
            p[n] *= dA;
        }
    }
    size_t o = (((size_t)b * NCHUNK + c) * D_HALF + d) * D_STATE;
    #pragma unroll
    for (int n = 0; n < D_STATE; ++n) { Pc[o + n] = p[n]; Xc[o + n] = x[n]; }
}

__global__ void scan_pass2(const float* __restrict__ Pc,
                           const float* __restrict__ Xc,
                           float* __restrict__ S)
{
    int idx = blockIdx.x * blockDim.x + threadIdx.x;
    if (idx >= B_SZ * D_HALF * D_STATE) return;
    int n = idx & (D_STATE - 1);
    int d = (idx >> 3) & (D_HALF - 1);
    int b = idx >> 11;
    float s = 0.f;
    for (int c = 0; c < NCHUNK; ++c) {
        size_t o = (((size_t)b * NCHUNK + c) * D_HALF + d) * D_STATE + n;
        S[o] = s;
        s = Pc[o] * s + Xc[o];
    }
}

__global__ void scan_pass3(const float* __restrict__ Delta,
                           const float* __restrict__ Hs,
                           const float* __restrict__ Xdbl,
                           const float* __restrict__ A_log,
                           const float* __restrict__ Dparam,
                           const float* __restrict__ S,
                           float* __restrict__ Z)
{
    int idx = blockIdx.x * blockDim.x + threadIdx.x;
    if (idx >= B_SZ * NCHUNK * D_HALF) return;
    int d = idx & (D_HALF - 1);
    int c = (idx >> 8) & (NCHUNK - 1);
    int b = idx >> 14;

    float a[D_STATE], x[D_STATE];
    size_t o = (((size_t)b * NCHUNK + c) * D_HALF + d) * D_STATE;
    #pragma unroll
    for (int n = 0; n < D_STATE; ++n) {
        a[n] = -expf(A_log[d * D_STATE + n]);
        x[n] = S[o + n];
    }
    float Dp = Dparam[d];
    int l0 = c * CHUNK;
    for (int t = 0; t < CHUNK; ++t) {
        size_t row = (size_t)b * L_SEQ + l0 + t;
        float dl = Delta[row * D_HALF + d];
        float u  = Hs[row * D_HALF + d];
        const float* bv  = Xdbl + row * XDBL_W + DT_RANK;
        const float* cvp = Xdbl + row * XDBL_W + DT_RANK + D_STATE;
        float du = dl * u;
        float y = 0.f;
        #pragma unroll
        for (int n = 0; n < D_STATE; ++n) {
            float dA = expf(dl * a[n]);
            x[n] = dA * x[n] + du * bv[n];
            y += cvp[n] * x[n];
        }
        Z[row * D_MODEL + D_HALF + d] = y + u * Dp;
    }
}

// ---------------------------------------------------------------------------
extern "C" void kernel_launch(void* const* d_in, const int* in_sizes, int n_in,
                              void* d_out, int out_size, void* d_ws, size_t ws_size,
                              hipStream_t stream)
{
    (void)in_sizes; (void)n_in; (void)out_size; (void)ws_size;
    const float* xy        = (const float*)d_in[0];
    const float* in_proj_w = (const float*)d_in[1];
    const float* conv_v_w  = (const float*)d_in[2];
    const float* conv_hs_w = (const float*)d_in[3];
    const float* hs_proj_w = (const float*)d_in[4];
    const float* dt_proj_w = (const float*)d_in[5];
    const float* dt_proj_b = (const float*)d_in[6];
    const float* A_log     = (const float*)d_in[7];
    const float* D_param   = (const float*)d_in[8];
    const float* out_proj_w= (const float*)d_in[9];
    float* out = (float*)d_out;

    // workspace layout (floats)
    float* P     = (float*)d_ws;                         // [16384,512]
    float* Z     = P     + (size_t)M_ROWS * D_MODEL;     // [16384,512] (v | y)
    float* Hs    = Z     + (size_t)M_ROWS * D_MODEL;     // [16384,256]
    float* Xdbl  = Hs    + (size_t)M_ROWS * D_HALF;      // [16384,48]
    float* Delta = Xdbl  + (size_t)M_ROWS * XDBL_W;      // [16384,256]
    float* Pc    = Delta + (size_t)M_ROWS * D_HALF;      // [4,64,256,8]
    float* Xc    = Pc    + (size_t)B_SZ * NCHUNK * D_HALF * D_STATE;
    float* Ssum  = Xc    + (size_t)B_SZ * NCHUNK * D_HALF * D_STATE;

    const int BLK = 256;

    // 1) in_proj: P = xy @ in_proj_w^T     (M=16384, N=512, K=512)
    gemm_f32_wmma<<<dim3((D_MODEL + BN - 1) / BN, M_ROWS / BM), BLK, 0, stream>>>(
        xy, D_MODEL, in_proj_w, D_MODEL, P, D_MODEL,
        M_ROWS, D_MODEL, D_MODEL, nullptr, 0);

    // 2) depthwise conv (SAME, k=3) + SiLU for both halves
    conv_silu<<<(B_SZ * L_SEQ * D_HALF + BLK - 1) / BLK, BLK, 0, stream>>>(
        P, conv_v_w, conv_hs_w, Z, Hs);

    // 3) hs_proj: Xdbl = Hs @ hs_proj_w^T   (N=48, K=256)
    gemm_f32_wmma<<<dim3((XDBL_W + BN - 1) / BN, M_ROWS / BM), BLK, 0, stream>>>(
        Hs, D_HALF, hs_proj_w, D_HALF, Xdbl, XDBL_W,
        M_ROWS, XDBL_W, D_HALF, nullptr, 0);

    // 4) dt-proj + fused double-bias softplus: Delta = softplus(Xdbl[:,:32]@W^T + 2b)
    gemm_f32_wmma<<<dim3((D_HALF + BN - 1) / BN, M_ROWS / BM), BLK, 0, stream>>>(
        Xdbl, XDBL_W, dt_proj_w, DT_RANK, Delta, D_HALF,
        M_ROWS, D_HALF, DT_RANK, dt_proj_b, 1);

    // 5) chunked selective scan (3 passes), y written into Z cols [256,512)
    scan_pass1<<<(B_SZ * NCHUNK * D_HALF + BLK - 1) / BLK, BLK, 0, stream>>>(
        Delta, Hs, Xdbl, A_log, Pc, Xc);
    scan_pass2<<<(B_SZ * D_HALF * D_STATE + BLK - 1) / BLK, BLK, 0, stream>>>(
        Pc, Xc, Ssum);
    scan_pass3<<<(B_SZ * NCHUNK * D_HALF + BLK - 1) / BLK, BLK, 0, stream>>>(
        Delta, Hs, Xdbl, A_log, D_param, Ssum, Z);

    // 6) out_proj: out = Z @ out_proj_w^T   (M=16384, N=512, K=512)
    gemm_f32_wmma<<<dim3((D_MODEL + BN - 1) / BN, M_ROWS / BM), BLK, 0, stream>>>(
        Z, D_MODEL, out_proj_w, D_MODEL, out, D_MODEL,
        M_ROWS, D_MODEL, D_MODEL, nullptr, 0);
}